// EsmFoldInvariantPointAttention_86827058856675
// MI455X (gfx1250) — compile-verified
//
#include <hip/hip_runtime.h>
#include <hip/hip_bf16.h>
#include <math.h>

#define NRES 512
#define CSD  1024
#define CZD  128
#define NH   12
#define HDD  16
#define PQN  4
#define PVN  8
#define CATD 2112   // H*(HD + PV*4 + CZ) = 12*(16+32+128) = 2112

typedef float v2f __attribute__((ext_vector_type(2)));
typedef float v8f __attribute__((ext_vector_type(8)));

// ---------------------------------------------------------------------------
// Generic f32 WMMA GEMM: C[b] = A[b](MxK) @ B[b](KxN) + bias
// One wave computes a 16 x (16*NT) tile: the A fragment is loaded once per
// k-step and reused across NT independent WMMAs (independent accumulators ->
// no RAW hazard chain on the matrix pipe).
// Requires: N multiple of 16*NT (grid covers exactly), K multiple of 4.
// A,B,C row-major. blockIdx.z = batch (strides in elements).
// ---------------------------------------------------------------------------
template <int NT>
__global__ void gemm_f32_wmma(const float* __restrict__ A, int lda, size_t strideA,
                              const float* __restrict__ Bm, int ldb, size_t strideB,
                              const float* __restrict__ bias,
                              float* __restrict__ C, int ldc, size_t strideC,
                              int K)
{
    const float* Ab = A + (size_t)blockIdx.z * strideA;
    const float* Bb = Bm + (size_t)blockIdx.z * strideB;
    float* Cb = C + (size_t)blockIdx.z * strideC;

    int lane = threadIdx.x & 31;
    int l16  = lane & 15;
    int half = lane >> 4;              // A/B: k += 2*half ; C: row += 8*half
    size_t row = (size_t)blockIdx.x * 16 + l16;
    int colBase = blockIdx.y * (16 * NT) + l16;

    v8f acc[NT];
#pragma unroll
    for (int t = 0; t < NT; ++t) acc[t] = (v8f){};

    for (int k0 = 0; k0 < K; k0 += 4) {
        int ka = k0 + 2 * half;
        v2f a;
        a.x = Ab[row * (size_t)lda + ka];
        a.y = Ab[row * (size_t)lda + ka + 1];
#pragma unroll
        for (int t = 0; t < NT; ++t) {
            v2f b;
            b.x = Bb[(size_t)ka * ldb + colBase + t * 16];
            b.y = Bb[(size_t)(ka + 1) * ldb + colBase + t * 16];
            acc[t] = __builtin_amdgcn_wmma_f32_16x16x4_f32(false, a, false, b,
                                                           (short)0, acc[t], false, false);
        }
    }
#pragma unroll
    for (int t = 0; t < NT; ++t) {
        int col = colBase + t * 16;
        float bv = bias ? bias[col] : 0.0f;
#pragma unroll
        for (int r = 0; r < 8; ++r) {
            size_t orow = (size_t)blockIdx.x * 16 + r + 8 * half;
            Cb[orow * (size_t)ldc + col] = acc[t][r] + bv;
        }
    }
}

// Pad Wb (128x12) -> (128x16), bb (12) -> (16)
__global__ void k_pad_wb(const float* __restrict__ Wb, const float* __restrict__ bb,
                         float* __restrict__ wbpad, float* __restrict__ bbpad)
{
    int t = blockIdx.x * blockDim.x + threadIdx.x;
    if (t < CZD * 16) {
        int r = t >> 4, c = t & 15;
        wbpad[t] = (c < NH) ? Wb[r * NH + c] : 0.0f;
    }
    if (t < 16) bbpad[t] = (t < NH) ? bb[t] : 0.0f;
}

// Rotate+translate raw points into global frame.
__global__ void k_points(const float* __restrict__ qpraw, const float* __restrict__ kvpraw,
                         const float* __restrict__ rot, const float* __restrict__ trans,
                         float* __restrict__ qpts, float* __restrict__ kpts,
                         float* __restrict__ vpts)
{
    int t = blockIdx.x * blockDim.x + threadIdx.x;
    if (t >= NRES * 192) return;
    int i = t / 192, slot = t % 192;
    const float* R = rot + (size_t)i * 9;
    const float* T = trans + (size_t)i * 3;
    float x, y, zc;
    if (slot < 48) {
        x  = qpraw[(size_t)i * 144 + slot];
        y  = qpraw[(size_t)i * 144 + 48 + slot];
        zc = qpraw[(size_t)i * 144 + 96 + slot];
    } else {
        int s2 = slot - 48;
        x  = kvpraw[(size_t)i * 432 + s2];
        y  = kvpraw[(size_t)i * 432 + 144 + s2];
        zc = kvpraw[(size_t)i * 432 + 288 + s2];
    }
    float o[3];
#pragma unroll
    for (int c = 0; c < 3; ++c)
        o[c] = R[c * 3 + 0] * x + R[c * 3 + 1] * y + R[c * 3 + 2] * zc + T[c];

    if (slot < 48) {
        int h = slot >> 2, p = slot & 3;
        float* dst = qpts + (((size_t)i * NH + h) * PQN + p) * 3;
        dst[0] = o[0]; dst[1] = o[1]; dst[2] = o[2];
    } else {
        int s2 = slot - 48;
        int h = s2 / 12, tt = s2 % 12;
        float* dst = (tt < PQN)
            ? kpts + (((size_t)i * NH + h) * PQN + tt) * 3
            : vpts + (((size_t)i * NH + h) * PVN + (tt - PQN)) * 3;
        dst[0] = o[0]; dst[1] = o[1]; dst[2] = o[2];
    }
}

// Build fused 32-dim attention vectors Qf/Kf per (i,h).
// logit = Qf.Kf  realizes  q.k*sqrt(1/48) + hw*sum(qp.kp) - 0.5*hw*(|qp|^2+|kp|^2)
__global__ void k_build_qkf(const float* __restrict__ qbuf, const float* __restrict__ kvbuf,
                            const float* __restrict__ qpts, const float* __restrict__ kpts,
                            const float* __restrict__ head_w,
                            float* __restrict__ Qf, float* __restrict__ Kf)
{
    int t = blockIdx.x * blockDim.x + threadIdx.x;
    if (t >= NRES * NH) return;
    int h = t % NH, i = t / NH;
    float hwraw = head_w[h];
    float sp = (hwraw > 20.0f) ? hwraw : log1pf(expf(hwraw));
    float hw = sp * 0.13608276348795434f;          // * sqrt(1/54)
    const float s1 = 0.14433756729740643f;          // sqrt(1/48)

    float* q = Qf + (size_t)i * (NH * 32) + h * 32;
    float* k = Kf + (size_t)i * (NH * 32) + h * 32;
#pragma unroll
    for (int d = 0; d < HDD; ++d) {
        q[d] = qbuf[(size_t)i * 192 + h * HDD + d] * s1;
        k[d] = kvbuf[(size_t)i * 384 + h * 32 + d];
    }
    float sq = 0.0f, sk = 0.0f;
#pragma unroll
    for (int p = 0; p < PQN; ++p)
#pragma unroll
        for (int c = 0; c < 3; ++c) {
            float qp = qpts[(((size_t)i * NH + h) * PQN + p) * 3 + c];
            float kp = kpts[(((size_t)i * NH + h) * PQN + p) * 3 + c];
            q[16 + p * 3 + c] = qp * hw;
            k[16 + p * 3 + c] = kp;
            sq += qp * qp;
            sk += kp * kp;
        }
    q[28] = -0.5f * hw * sq;  k[28] = 1.0f;
    q[29] = 1.0f;             k[29] = -0.5f * hw * sk;
    q[30] = 0.0f; q[31] = 0.0f; k[30] = 0.0f; k[31] = 0.0f;
}

// Logits per head: wave computes a 16 x 64 tile (4 J-tiles) with K=32.
// Qf fragment loaded once per k-step, reused across 4 WMMAs.
__global__ void k_logits(const float* __restrict__ Qf, const float* __restrict__ Kf,
                         const float* __restrict__ bpad, const float* __restrict__ mask,
                         float* __restrict__ logits)
{
    int h = blockIdx.z;
    int lane = threadIdx.x & 31;
    int l16 = lane & 15, half = lane >> 4;
    int i0 = blockIdx.x * 16, j0 = blockIdx.y * 64;
    int irow = i0 + l16;

    v8f acc[4];
#pragma unroll
    for (int t = 0; t < 4; ++t) acc[t] = (v8f){};

#pragma unroll
    for (int k0 = 0; k0 < 32; k0 += 4) {
        int ka = k0 + 2 * half;
        v2f a;
        a.x = Qf[(size_t)irow * (NH * 32) + h * 32 + ka];
        a.y = Qf[(size_t)irow * (NH * 32) + h * 32 + ka + 1];
#pragma unroll
        for (int t = 0; t < 4; ++t) {
            int jcol = j0 + t * 16 + l16;
            v2f b;
            b.x = Kf[(size_t)jcol * (NH * 32) + h * 32 + ka];
            b.y = Kf[(size_t)jcol * (NH * 32) + h * 32 + ka + 1];
            acc[t] = __builtin_amdgcn_wmma_f32_16x16x4_f32(false, a, false, b,
                                                           (short)0, acc[t], false, false);
        }
    }
#pragma unroll
    for (int t = 0; t < 4; ++t) {
        int jcol = j0 + t * 16 + l16;
        float mj = mask[jcol];
#pragma unroll
        for (int r = 0; r < 8; ++r) {
            int i = i0 + r + 8 * half;
            float v = acc[t][r]
                    + 0.5773502691896258f * bpad[((size_t)i * NRES + jcol) * 16 + h]
                    + 100000.0f * (mask[i] * mj - 1.0f);
            logits[((size_t)h * NRES + i) * NRES + jcol] = v;
        }
    }
}

// In-place softmax over last dim (512) of (H,N,N); one wave per row.
__global__ void k_softmax(float* __restrict__ a)
{
    int row = blockIdx.x * (blockDim.x >> 5) + (threadIdx.x >> 5);
    int lane = threadIdx.x & 31;
    float* p = a + (size_t)row * NRES;
    float m = -1e30f;
    for (int j = lane; j < NRES; j += 32) m = fmaxf(m, p[j]);
#pragma unroll
    for (int o = 16; o > 0; o >>= 1) m = fmaxf(m, __shfl_xor(m, o, 32));
    float s = 0.0f;
    for (int j = lane; j < NRES; j += 32) { float e = __expf(p[j] - m); p[j] = e; s += e; }
#pragma unroll
    for (int o = 16; o > 0; o >>= 1) s += __shfl_xor(s, o, 32);
    float inv = 1.0f / s;
    for (int j = lane; j < NRES; j += 32) p[j] *= inv;
}

// Pack per-head value matrix: [v(16) | v_pts flat(24) | pad(8)] -> (H,N,48)
__global__ void k_pack_v(const float* __restrict__ kvbuf, const float* __restrict__ vpts,
                         float* __restrict__ vpack)
{
    int t = blockIdx.x * blockDim.x + threadIdx.x;
    if (t >= NH * NRES * 48) return;
    int c = t % 48, j = (t / 48) % NRES, h = t / (48 * NRES);
    float v;
    if (c < 16)      v = kvbuf[(size_t)j * 384 + h * 32 + 16 + c];
    else if (c < 40) { int pc = c - 16; v = vpts[(((size_t)j * NH + h) * PVN + pc / 3) * 3 + pc % 3]; }
    else             v = 0.0f;
    vpack[t] = v;
}

// o_pair[i,h,c] = sum_j a[h,i,j] * z[i,j,c]
// Wave computes 16(pad12) x 64 tile (4 channel tiles), K=512; the attention
// fragment is loaded once per k-step.
__global__ void k_opair(const float* __restrict__ a, const float* __restrict__ z,
                        float* __restrict__ opair)
{
    int i = blockIdx.x;
    int c0 = blockIdx.y * 64;
    int lane = threadIdx.x & 31, l16 = lane & 15, half = lane >> 4;
    int m = l16;
    bool mok = (m < NH);

    v8f acc[4];
#pragma unroll
    for (int t = 0; t < 4; ++t) acc[t] = (v8f){};

    for (int k0 = 0; k0 < NRES; k0 += 4) {
        int ka = k0 + 2 * half;
        v2f av;
        av.x = mok ? a[((size_t)m * NRES + i) * NRES + ka] : 0.0f;
        av.y = mok ? a[((size_t)m * NRES + i) * NRES + ka + 1] : 0.0f;
#pragma unroll
        for (int t = 0; t < 4; ++t) {
            v2f bv;
            bv.x = z[((size_t)i * NRES + ka) * CZD + c0 + t * 16 + l16];
            bv.y = z[((size_t)i * NRES + ka + 1) * CZD + c0 + t * 16 + l16];
            acc[t] = __builtin_amdgcn_wmma_f32_16x16x4_f32(false, av, false, bv,
                                                           (short)0, acc[t], false, false);
        }
    }
#pragma unroll
    for (int t = 0; t < 4; ++t) {
#pragma unroll
        for (int r = 0; r < 8; ++r) {
            int h = r + 8 * half;
            if (h < NH)
                opair[((size_t)i * NH + h) * CZD + c0 + t * 16 + l16] = acc[t][r];
        }
    }
}

// Inverse frame: loc = R^T (o_pt - t), plus norms.
__global__ void k_ptinv(const float* __restrict__ opack, const float* __restrict__ rot,
                        const float* __restrict__ trans,
                        float* __restrict__ optloc, float* __restrict__ norms)
{
    int t = blockIdx.x * blockDim.x + threadIdx.x;
    if (t >= NRES * NH * PVN) return;
    int p = t % PVN, h = (t / PVN) % NH, i = t / (PVN * NH);
    const float* R = rot + (size_t)i * 9;
    float d[3];
#pragma unroll
    for (int c = 0; c < 3; ++c)
        d[c] = opack[((size_t)h * NRES + i) * 48 + 16 + p * 3 + c] - trans[(size_t)i * 3 + c];
    float n2 = 1e-8f;
#pragma unroll
    for (int c = 0; c < 3; ++c) {
        float loc = R[0 * 3 + c] * d[0] + R[1 * 3 + c] * d[1] + R[2 * 3 + c] * d[2];
        optloc[(size_t)t * 3 + c] = loc;
        n2 += loc * loc;
    }
    norms[t] = sqrtf(n2);
}

// Assemble cat(i, 2112) = [o | pt.x | pt.y | pt.z | ptnorm | o_pair]
__global__ void k_build_cat(const float* __restrict__ opack, const float* __restrict__ optloc,
                            const float* __restrict__ norms, const float* __restrict__ opair,
                            float* __restrict__ cat)
{
    int t = blockIdx.x * blockDim.x + threadIdx.x;
    if (t >= NRES * CATD) return;
    int i = t / CATD, col = t % CATD;
    float v;
    if (col < 192) {
        int h = col >> 4, d = col & 15;
        v = opack[((size_t)h * NRES + i) * 48 + d];
    } else if (col < 480) {
        int comp = (col - 192) / 96;
        int idx  = (col - 192) % 96;                 // h*8+p
        v = optloc[((size_t)i * 96 + idx) * 3 + comp];
    } else if (col < 576) {
        v = norms[(size_t)i * 96 + (col - 480)];
    } else {
        v = opair[(size_t)i * (NH * CZD) + (col - 576)];
    }
    cat[(size_t)i * CATD + col] = v;
}

extern "C" void kernel_launch(void* const* d_in, const int* in_sizes, int n_in,
                              void* d_out, int out_size, void* d_ws, size_t ws_size,
                              hipStream_t stream)
{
    const float* s      = (const float*)d_in[0];
    const float* z      = (const float*)d_in[1];
    const float* rot    = (const float*)d_in[2];
    const float* trans  = (const float*)d_in[3];
    const float* mask   = (const float*)d_in[4];
    const float* Wq     = (const float*)d_in[5];
    const float* bq     = (const float*)d_in[6];
    const float* Wkv    = (const float*)d_in[7];
    const float* bkv    = (const float*)d_in[8];
    const float* Wqp    = (const float*)d_in[9];
    const float* bqp    = (const float*)d_in[10];
    const float* Wkvp   = (const float*)d_in[11];
    const float* bkvp   = (const float*)d_in[12];
    const float* Wb     = (const float*)d_in[13];
    const float* bb     = (const float*)d_in[14];
    const float* head_w = (const float*)d_in[15];
    const float* Wout   = (const float*)d_in[16];
    const float* bout   = (const float*)d_in[17];
    float* out = (float*)d_out;

    float* ws = (float*)d_ws;
    size_t off = 0;
    auto alloc = [&](size_t n) { float* p = ws + off; off += (n + 63) & ~(size_t)63; return p; };

    float* qbuf   = alloc((size_t)NRES * 192);        // q projection
    float* kvbuf  = alloc((size_t)NRES * 384);        // k|v per head
    float* qpraw  = alloc((size_t)NRES * 144);
    float* kvpraw = alloc((size_t)NRES * 432);
    float* qpts   = alloc((size_t)NRES * NH * PQN * 3);
    float* kpts   = alloc((size_t)NRES * NH * PQN * 3);
    float* vpts   = alloc((size_t)NRES * NH * PVN * 3);
    float* Qf     = alloc((size_t)NRES * NH * 32);
    float* Kf     = alloc((size_t)NRES * NH * 32);
    float* wbpad  = alloc((size_t)CZD * 16);
    float* bbpad  = alloc(16);
    float* bpad   = alloc((size_t)NRES * NRES * 16);  // pair bias, padded heads
    float* logits = alloc((size_t)NH * NRES * NRES);  // becomes softmax `a` in place
    float* vpack  = alloc((size_t)NH * NRES * 48);
    float* opack  = alloc((size_t)NH * NRES * 48);    // [o(16)|o_pt global(24)|pad]
    float* opair  = alloc((size_t)NRES * NH * CZD);
    float* optloc = alloc((size_t)NRES * NH * PVN * 3);
    float* norms  = alloc((size_t)NRES * NH * PVN);
    float* cat    = alloc((size_t)NRES * CATD);

    dim3 w32(32);

    // Projections from s (K=1024)
    gemm_f32_wmma<4><<<dim3(32, 3, 1), w32, 0, stream>>>(s, CSD, 0, Wq,   192, 0, bq,   qbuf,   192, 0, CSD);
    gemm_f32_wmma<4><<<dim3(32, 6, 1), w32, 0, stream>>>(s, CSD, 0, Wkv,  384, 0, bkv,  kvbuf,  384, 0, CSD);
    gemm_f32_wmma<3><<<dim3(32, 3, 1), w32, 0, stream>>>(s, CSD, 0, Wqp,  144, 0, bqp,  qpraw,  144, 0, CSD);
    gemm_f32_wmma<3><<<dim3(32, 9, 1), w32, 0, stream>>>(s, CSD, 0, Wkvp, 432, 0, bkvp, kvpraw, 432, 0, CSD);

    // Pair bias: (N*N x 128) @ (128 x 16pad)
    k_pad_wb<<<dim3(8), dim3(256), 0, stream>>>(Wb, bb, wbpad, bbpad);
    gemm_f32_wmma<1><<<dim3(NRES * NRES / 16, 1, 1), w32, 0, stream>>>(
        z, CZD, 0, wbpad, 16, 0, bbpad, bpad, 16, 0, CZD);

    // Points to global frame; fused attention vectors
    k_points<<<dim3((NRES * 192 + 255) / 256), dim3(256), 0, stream>>>(
        qpraw, kvpraw, rot, trans, qpts, kpts, vpts);
    k_build_qkf<<<dim3((NRES * NH + 255) / 256), dim3(256), 0, stream>>>(
        qbuf, kvbuf, qpts, kpts, head_w, Qf, Kf);

    // Logits + softmax
    k_logits<<<dim3(32, 8, NH), w32, 0, stream>>>(Qf, Kf, bpad, mask, logits);
    k_softmax<<<dim3(NH * NRES / 8), dim3(256), 0, stream>>>(logits);

    // o and o_pt (global): batched per-head GEMM a(512x512) @ vpack(512x48)
    k_pack_v<<<dim3((NH * NRES * 48 + 255) / 256), dim3(256), 0, stream>>>(kvbuf, vpts, vpack);
    gemm_f32_wmma<3><<<dim3(32, 1, NH), w32, 0, stream>>>(
        logits, NRES, (size_t)NRES * NRES,
        vpack, 48, (size_t)NRES * 48,
        nullptr,
        opack, 48, (size_t)NRES * 48, NRES);

    // o_pair (per residue), inverse frame, cat
    k_opair<<<dim3(NRES, CZD / 64, 1), w32, 0, stream>>>(logits, z, opair);
    k_ptinv<<<dim3((NRES * NH * PVN + 255) / 256), dim3(256), 0, stream>>>(
        opack, rot, trans, optloc, norms);
    k_build_cat<<<dim3((NRES * CATD + 255) / 256), dim3(256), 0, stream>>>(
        opack, optloc, norms, opair, cat);

    // Final projection: cat(512x2112) @ Wout(2112x1024) + bout
    gemm_f32_wmma<4><<<dim3(32, 16, 1), w32, 0, stream>>>(
        cat, CATD, 0, Wout, CSD, 0, bout, out, CSD, 0, CATD);
}